// ResidualVectorQuantizer_65987877536152
// MI455X (gfx1250) — compile-verified
//
#include <hip/hip_runtime.h>

typedef float v2f __attribute__((ext_vector_type(2)));
typedef float v8f __attribute__((ext_vector_type(8)));

#define B_ 8
#define T_ 32768
#define D_ 8
#define Q_ 8
#define K_ 1024
#define NT (B_ * T_)            // 262144 tokens
#define THREADS 256             // 8 waves (wave32)
#define TOK_PER_BLOCK 256       // 16 tokens per wave
#define NBLOCKS (NT / TOK_PER_BLOCK)
#define CB_STRIDE 10            // 8 data floats + cnorm + pad (8B-aligned pairs)

// One kernel does all 8 RVQ stages with the residual held in registers.
// Cross-terms r.c via V_WMMA_F32_16X16X4_F32 (two K=4 steps for D=8).
__global__ __launch_bounds__(THREADS)
void rvq_stage_kernel(const float* __restrict__ x,
                      const float* __restrict__ cb,      // [Q,K,D]
                      float* __restrict__ qout,          // [B,T,D]
                      float* __restrict__ codes_out,     // [Q,B,T] as float
                      float* __restrict__ loss_partials) // [NBLOCKS]
{
    __shared__ float cb_lds[K_ * CB_STRIDE];   // 40 KB: codebook + cnorm per code
    __shared__ int   idx_lds[TOK_PER_BLOCK];   // per-token argmin exchange
    __shared__ float wsum[THREADS / 32];       // per-wave loss partials

    const int tid  = threadIdx.x;
    const int wave = tid >> 5;
    const int lane = tid & 31;
    const int half = lane >> 4;     // 0: k={0,1,4,5}; 1: k={2,3,6,7}
    const int l16  = lane & 15;

    const int row_tok = blockIdx.x * TOK_PER_BLOCK + wave * 16 + l16;
    const int klo = 2 * half;       // first k-pair for this lane (A/B frag layout)
    const int khi = 4 + 2 * half;   // second k-pair

    // x fragment == initial residual, in WMMA-A fragment layout
    const float* xrow = x + (size_t)row_tok * D_;
    v2f xlo = *(const v2f*)(xrow + klo);
    v2f xhi = *(const v2f*)(xrow + khi);
    v2f rlo = xlo, rhi = xhi;

    float lossacc = 0.0f;

    for (int q = 0; q < Q_; ++q) {
        // ---- stage codebook -> LDS, with cnorm precomputed into slot 8 ----
        __syncthreads();
        const float4* gsrc = (const float4*)(cb + (size_t)q * K_ * D_);
        #pragma unroll
        for (int j = 0; j < K_ / THREADS; ++j) {
            int c = tid + j * THREADS;
            float4 p0 = gsrc[c * 2 + 0];
            float4 p1 = gsrc[c * 2 + 1];
            float* dst = &cb_lds[c * CB_STRIDE];
            dst[0] = p0.x; dst[1] = p0.y; dst[2] = p0.z; dst[3] = p0.w;
            dst[4] = p1.x; dst[5] = p1.y; dst[6] = p1.z; dst[7] = p1.w;
            dst[8] = p0.x*p0.x + p0.y*p0.y + p0.z*p0.z + p0.w*p0.w
                   + p1.x*p1.x + p1.y*p1.y + p1.z*p1.z + p1.w*p1.w;
        }
        __syncthreads();

        if (q + 1 < Q_)  // warm L2 for next stage (global_prefetch_b8)
            __builtin_prefetch(cb + (size_t)(q + 1) * K_ * D_ + tid * D_, 0, 1);

        // ---- argmin over 1024 codes; ||r||^2 is row-constant -> dropped ----
        v8f best_d;
        int best_i[8];
        #pragma unroll
        for (int v = 0; v < 8; ++v) { best_d[v] = 3.4e38f; best_i[v] = 0; }

        const v2f alo = rlo, ahi = rhi;

        for (int n0 = 0; n0 < K_; n0 += 16) {
            const int n = n0 + l16;                    // this lane's code column
            const float* crow = &cb_lds[n * CB_STRIDE];
            v2f  blo = *(const v2f*)(crow + klo);      // B-frag k-pair (K=0..3)
            v2f  bhi = *(const v2f*)(crow + khi);      // B-frag k-pair (K=4..7)
            float cn = crow[8];

            v8f acc = {};
            acc = __builtin_amdgcn_wmma_f32_16x16x4_f32(false, alo, false, blo,
                                                        (short)0, acc, false, false);
            acc = __builtin_amdgcn_wmma_f32_16x16x4_f32(false, ahi, false, bhi,
                                                        (short)0, acc, false, false);
            #pragma unroll
            for (int v = 0; v < 8; ++v) {
                float d = __builtin_fmaf(acc[v], -2.0f, cn);  // cnorm - 2*r.c
                bool t = d < best_d[v];                       // strict: first-min
                best_d[v] = t ? d : best_d[v];
                best_i[v] = t ? n : best_i[v];
            }
        }

        // butterfly argmin across the 16 code-lanes of each half (tie -> low idx)
        #pragma unroll
        for (int m = 8; m >= 1; m >>= 1) {
            #pragma unroll
            for (int v = 0; v < 8; ++v) {
                float od = __shfl_xor(best_d[v], m, 16);
                int   oi = __shfl_xor(best_i[v], m, 16);
                bool t = (od < best_d[v]) || (od == best_d[v] && oi < best_i[v]);
                best_d[v] = t ? od : best_d[v];
                best_i[v] = t ? oi : best_i[v];
            }
        }

        // rows for half h live in vgprs 0..7 as rows 8h..8h+7; publish via LDS
        if (l16 == 0) {
            #pragma unroll
            for (int v = 0; v < 8; ++v)
                idx_lds[wave * 16 + half * 8 + v] = best_i[v];
        }
        __syncthreads();

        const int myidx = idx_lds[wave * 16 + l16];
        if (half == 0)
            codes_out[(size_t)q * NT + row_tok] = (float)myidx;

        // gather embedding fragment, update residual, accumulate loss
        const float* erow = &cb_lds[myidx * CB_STRIDE];
        v2f elo = *(const v2f*)(erow + klo);
        v2f ehi = *(const v2f*)(erow + khi);
        rlo -= elo; rhi -= ehi;                        // r_new = r - emb
        v2f dlo = rlo - elo, dhi = rhi - ehi;          // (r_new - emb)
        lossacc += dlo.x*dlo.x + dlo.y*dlo.y + dhi.x*dhi.x + dhi.y*dhi.y;
    }

    // quantized = x - final residual (value path of straight-through estimator)
    float* qrow = qout + (size_t)row_tok * D_;
    *(v2f*)(qrow + klo) = xlo - rlo;
    *(v2f*)(qrow + khi) = xhi - rhi;

    // loss: wave reduce -> block reduce -> per-block partial (deterministic)
    #pragma unroll
    for (int m = 16; m >= 1; m >>= 1)
        lossacc += __shfl_xor(lossacc, m, 32);
    if (lane == 0) wsum[wave] = lossacc;
    __syncthreads();
    if (tid == 0) {
        float s = 0.0f;
        #pragma unroll
        for (int w = 0; w < THREADS / 32; ++w) s += wsum[w];
        loss_partials[blockIdx.x] = s;
    }
}

// Fixed-order partial sum -> fully deterministic scalar loss.
__global__ void rvq_loss_reduce(const float* __restrict__ partials,
                                float* __restrict__ loss_out)
{
    if (threadIdx.x == 0 && blockIdx.x == 0) {
        float s = 0.0f;
        for (int i = 0; i < NBLOCKS; ++i) s += partials[i];
        *loss_out = s * (0.25f / (float)((size_t)NT * D_));
    }
}

extern "C" void kernel_launch(void* const* d_in, const int* in_sizes, int n_in,
                              void* d_out, int out_size, void* d_ws, size_t ws_size,
                              hipStream_t stream) {
    const float* x  = (const float*)d_in[0];   // [8, 32768, 8]
    const float* cb = (const float*)d_in[1];   // [8, 1024, 8]

    float* qout  = (float*)d_out;              // [B,T,D]
    float* loss  = qout + (size_t)NT * D_;     // scalar
    float* codes = loss + 1;                   // [Q,B,T] as float
    float* partials = (float*)d_ws;            // NBLOCKS floats

    rvq_stage_kernel<<<NBLOCKS, THREADS, 0, stream>>>(x, cb, qout, codes, partials);
    rvq_loss_reduce<<<1, 32, 0, stream>>>(partials, loss);
}